// CRF_13915694039459
// MI455X (gfx1250) — compile-verified
//
#include <hip/hip_runtime.h>
#include <cstdint>

// ---------------- problem constants ----------------
#define K        16
#define T_TOTAL  1000000
#define CHUNK    2000
#define NCHUNK   500            // CHUNK * NCHUNK == T_TOTAL
#define TILE     125            // steps staged to LDS at a time
#define NTILE    (CHUNK / TILE) // 16
#define NEG_INF  (-3.0e30f)

// ---------------- workspace layout (bytes) ----------------
#define WS_A     0                          // NCHUNK * 256 floats  (chunk max-plus matrices)
#define WS_BFV   (WS_A    + NCHUNK*256*4)   // NCHUNK * 16 floats   (fv entering each chunk)
#define WS_G     (WS_BFV  + NCHUNK*16*4)    // NCHUNK * 16 bytes    (exit-tag -> entry-tag maps)
#define WS_EXIT  (WS_G    + NCHUNK*16)      // NCHUNK ints          (exit tag per chunk)
#define WS_LAST  (WS_EXIT + NCHUNK*4)       // 1 int                (global argmax tag)

// ---------------- CDNA5 async global -> LDS (inline asm; ASYNCcnt path) ----------------
__device__ __forceinline__ void async_copy_b128(void* ldsDst, const void* gSrc) {
  uint32_t l = (uint32_t)(uintptr_t)ldsDst;     // low 32 bits of generic shared ptr = LDS offset
  asm volatile("global_load_async_to_lds_b128 %0, %1, off"
               :
               : "v"(l), "v"(gSrc)
               : "memory");
}
__device__ __forceinline__ void wait_async() {
  asm volatile("s_wait_asynccnt 0" ::: "memory");
}

// ---------------- CDNA5 Tensor Data Mover (TENSORcnt path) ----------------
#if defined(__AMDGCN__) && __has_builtin(__builtin_amdgcn_tensor_load_to_lds)
#define HAVE_TDM 1
typedef unsigned int tdm_u32x4 __attribute__((ext_vector_type(4)));
typedef int          tdm_i32x4 __attribute__((ext_vector_type(4)));
typedef int          tdm_i32x8 __attribute__((ext_vector_type(8)));

// 1-D tile: nElemF32 consecutive f32 from global -> LDS.  D# per CDNA5 ISA:
// group0 = {count=1 | lds_addr | global_addr[31:0] | addr[56:32]|type=2<<30}
// group1 = {data_size=4B<<16 | tensor_dim0<<16 | 0 | tile_dim0<<16 | 0...}
__device__ __forceinline__ void tdm_load_tile(uint32_t ldsAddr, const void* gSrc,
                                              uint32_t nElemF32) {
  uintptr_t ga = (uintptr_t)gSrc;
  tdm_u32x4 g0 = { 1u,
                   ldsAddr,
                   (uint32_t)ga,
                   (uint32_t)((ga >> 32) & 0x1FFFFFFu) | (2u << 30) };
  tdm_i32x8 g1 = { (int)(2u << 16),          // data_size = 4 bytes
                   (int)(nElemF32 << 16),    // tensor_dim0[15:0] (bits 79:48 low half)
                   0,                        // tensor_dim0 hi / tensor_dim1 lo
                   (int)(nElemF32 << 16),    // tile_dim0 (bits 127:112)
                   0, 0, 0, 0 };             // tile_dim1/2 = 0 (1-D), strides = 0
  tdm_i32x4 gz = { 0, 0, 0, 0 };
#if __clang_major__ >= 23
  tdm_i32x8 gz8 = { 0, 0, 0, 0, 0, 0, 0, 0 };
  __builtin_amdgcn_tensor_load_to_lds(g0, g1, gz, gz, gz8, 0);
#else
  __builtin_amdgcn_tensor_load_to_lds(g0, g1, gz, gz, 0);
#endif
}
#endif

// =====================================================================
// K1: per-chunk 16x16 max-plus transfer matrix.
//     block = 256 = 16 columns x 16 next-tags; one chunk per block.
//     Feat tiles staged by the Tensor Data Mover (one wave issues,
//     s_wait_tensorcnt 0, then block barrier publishes the tile).
// =====================================================================
__global__ __launch_bounds__(256) void viterbi_chunk_mats(
    const float* __restrict__ sent, const float* __restrict__ trans,
    float* __restrict__ A)
{
  __shared__ alignas(16) float featLds[TILE * K];
  __shared__ alignas(16) float exch[256];
  const int tid = threadIdx.x;
  const int c   = blockIdx.x;
  const int n   = tid & 15;
  const int q   = tid >> 4;

  float tr[K];
#pragma unroll
  for (int p = 0; p < K; ++p) tr[p] = trans[n * K + p];

  float cur = (n == q) ? 0.0f : NEG_INF;            // unit column init in (max,+)
  const float4* fg4 = (const float4*)(exch + (tid & ~15));  // 64B-aligned window

  for (int tile = 0; tile < NTILE; ++tile) {
    __syncthreads();
    const float* src = sent + (size_t)(c * CHUNK + tile * TILE) * K;
#if defined(HAVE_TDM)
    if (tid < 32) {                                 // one wave drives the TDM
      tdm_load_tile((uint32_t)(uintptr_t)featLds, src, TILE * K);
      __builtin_amdgcn_s_wait_tensorcnt(0);
    }
#else
    for (int i = tid; i < (TILE * K) / 4; i += 256)
      async_copy_b128(&featLds[i * 4], (const float4*)src + i);
    wait_async();
#endif
    __syncthreads();

    for (int s = 0; s < TILE; ++s) {
      exch[tid] = cur;
      __builtin_amdgcn_wave_barrier();
      float4 e0 = fg4[0], e1 = fg4[1], e2 = fg4[2], e3 = fg4[3];   // ds_load_b128 x4
      float fv[K] = { e0.x, e0.y, e0.z, e0.w,  e1.x, e1.y, e1.z, e1.w,
                      e2.x, e2.y, e2.z, e2.w,  e3.x, e3.y, e3.z, e3.w };
      float best = fv[0] + tr[0];
#pragma unroll
      for (int p = 1; p < K; ++p) best = fmaxf(best, fv[p] + tr[p]);
      cur = best + featLds[s * K + n];
      __builtin_amdgcn_wave_barrier();
    }
  }
  A[(size_t)c * 256 + n * K + q] = cur;             // A[c][n][q]
}

// =====================================================================
// K2: sequential max-plus scan over the 500 chunk matrices (1 wave).
// =====================================================================
__global__ __launch_bounds__(32) void viterbi_scan_fwd(
    const float* __restrict__ A, float* __restrict__ bfv, int* __restrict__ lastTag)
{
  __shared__ alignas(16) float fvb[K];
  const int tid = threadIdx.x;
  if (tid < K) {
    const int n = tid;
    fvb[n] = 0.0f;                                  // reference init_fv = zeros
    __builtin_amdgcn_wave_barrier();
    const float4* fb4 = (const float4*)fvb;
    for (int c = 0; c < NCHUNK; ++c) {
      bfv[c * K + n] = fvb[n];
      const float* ar = A + (size_t)c * 256 + n * K;
      if (c + 1 < NCHUNK)
        __builtin_prefetch(A + (size_t)(c + 1) * 256 + n * K, 0, 0);
      float4 e0 = fb4[0], e1 = fb4[1], e2 = fb4[2], e3 = fb4[3];
      float fv[K] = { e0.x, e0.y, e0.z, e0.w,  e1.x, e1.y, e1.z, e1.w,
                      e2.x, e2.y, e2.z, e2.w,  e3.x, e3.y, e3.z, e3.w };
      float best = ar[0] + fv[0];
#pragma unroll
      for (int qq = 1; qq < K; ++qq) best = fmaxf(best, ar[qq] + fv[qq]);
      __builtin_amdgcn_wave_barrier();
      fvb[n] = best;
      __builtin_amdgcn_wave_barrier();
    }
    if (n == 0) {
      float best = fvb[0]; int bi = 0;
      for (int j = 1; j < K; ++j) { if (fvb[j] > best) { best = fvb[j]; bi = j; } }
      *lastTag = bi;
    }
  }
}

// =====================================================================
// Shared recursion body for K3/K5: rebuild one chunk's backpointers in
// LDS from its true boundary fv.  block = 32 (one wave), async-LDS
// staging (ASYNCcnt path).
// =====================================================================
__device__ __forceinline__ void chunk_recursion(
    const float* __restrict__ sent, const float* __restrict__ trans,
    const float* __restrict__ bfv, int c, int tid,
    float* featLds, unsigned char* bp, float* exch)
{
  const int n = tid & 15;
  float tr[K];
#pragma unroll
  for (int p = 0; p < K; ++p) tr[p] = trans[n * K + p];

  float cur = bfv[c * K + n];
  const float4* ex4 = (const float4*)exch;

  for (int tile = 0; tile < NTILE; ++tile) {
    __syncthreads();
    const float4* src = (const float4*)(sent + (size_t)(c * CHUNK + tile * TILE) * K);
    for (int i = tid; i < (TILE * K) / 4; i += 32)
      async_copy_b128(&featLds[i * 4], src + i);
    wait_async();
    __syncthreads();

    if (tid < K) {
      for (int s = 0; s < TILE; ++s) {
        exch[n] = cur;
        __builtin_amdgcn_wave_barrier();
        float4 e0 = ex4[0], e1 = ex4[1], e2 = ex4[2], e3 = ex4[3];
        float fv[K] = { e0.x, e0.y, e0.z, e0.w,  e1.x, e1.y, e1.z, e1.w,
                        e2.x, e2.y, e2.z, e2.w,  e3.x, e3.y, e3.z, e3.w };
        float best = fv[0] + tr[0]; int bi = 0;
#pragma unroll
        for (int p = 1; p < K; ++p) {
          float sc = fv[p] + tr[p];
          if (sc > best) { best = sc; bi = p; }     // strict '>' == jnp.argmax first-max
        }
        bp[(tile * TILE + s) * K + n] = (unsigned char)bi;
        cur = best + featLds[s * K + n];
        __builtin_amdgcn_wave_barrier();
      }
    }
  }
  __syncthreads();
}

// K3: recursion + chunk backtrack map G_c (exit tag -> entry tag).
__global__ __launch_bounds__(32) void viterbi_chunk_bp_g(
    const float* __restrict__ sent, const float* __restrict__ trans,
    const float* __restrict__ bfv, unsigned char* __restrict__ G)
{
  __shared__ alignas(16) float featLds[TILE * K];
  __shared__ alignas(16) unsigned char bp[CHUNK * K];
  __shared__ alignas(16) float exch[K];
  const int tid = threadIdx.x;
  const int c   = blockIdx.x;

  chunk_recursion(sent, trans, bfv, c, tid, featLds, bp, exch);

  if (tid < K) {
    int tag = tid;                                  // hypothetical exit tag
    for (int t = CHUNK - 1; t >= 0; --t) tag = bp[t * K + tag];
    G[c * K + tid] = (unsigned char)tag;            // entry tag reached
  }
}

// =====================================================================
// K4: reverse scan of the chunk backtrack maps (tiny; G staged in LDS).
// =====================================================================
__global__ __launch_bounds__(256) void viterbi_scan_bwd(
    const unsigned char* __restrict__ G, const int* __restrict__ lastTag,
    int* __restrict__ exitTag)
{
  __shared__ alignas(16) unsigned char Gl[NCHUNK * K];
  const int tid = threadIdx.x;
  for (int i = tid; i < (NCHUNK * K) / 4; i += 256)
    ((int*)Gl)[i] = ((const int*)G)[i];
  __syncthreads();
  if (tid == 0) {
    int e = *lastTag;
    for (int c = NCHUNK - 1; c >= 0; --c) {
      exitTag[c] = e;
      e = Gl[c * K + e];
    }
  }
}

// =====================================================================
// K5: recursion again (sentence re-read is L2-resident: 64 MB << 192 MB
//     L2), backtrack from the known exit tag into LDS, coalesced flush.
// =====================================================================
__global__ __launch_bounds__(32) void viterbi_emit(
    const float* __restrict__ sent, const float* __restrict__ trans,
    const float* __restrict__ bfv, const int* __restrict__ exitTag,
    int* __restrict__ out)
{
  __shared__ alignas(16) float featLds[TILE * K];
  __shared__ alignas(16) unsigned char bp[CHUNK * K];
  __shared__ alignas(16) float exch[K];
  __shared__ alignas(16) unsigned char pathB[CHUNK];
  const int tid = threadIdx.x;
  const int c   = blockIdx.x;

  chunk_recursion(sent, trans, bfv, c, tid, featLds, bp, exch);

  if (tid == 0) {
    int tag = exitTag[c];                           // tag after this chunk's last step
    for (int t = CHUNK - 1; t >= 0; --t) {
      int pv = bp[t * K + tag];
      pathB[t] = (unsigned char)pv;                 // best_path[t] = bp_t[tag_{t+1}]
      tag = pv;
    }
  }
  __syncthreads();
  for (int i = tid; i < CHUNK; i += 32)             // coalesced int32 emission
    out[(size_t)c * CHUNK + i] = (int)pathB[i];
}

// =====================================================================
extern "C" void kernel_launch(void* const* d_in, const int* in_sizes, int n_in,
                              void* d_out, int out_size, void* d_ws, size_t ws_size,
                              hipStream_t stream) {
  (void)in_sizes; (void)n_in; (void)out_size; (void)ws_size;
  const float* sent  = (const float*)d_in[0];   // [1, T, 16] f32
  const float* trans = (const float*)d_in[1];   // [16, 16]   f32
  int* out = (int*)d_out;                       // [T] int32 tag ids

  char* ws = (char*)d_ws;
  float*         A    = (float*)(ws + WS_A);
  float*         bfv  = (float*)(ws + WS_BFV);
  unsigned char* G    = (unsigned char*)(ws + WS_G);
  int*           exT  = (int*)(ws + WS_EXIT);
  int*           last = (int*)(ws + WS_LAST);

  viterbi_chunk_mats<<<NCHUNK, 256, 0, stream>>>(sent, trans, A);
  viterbi_scan_fwd  <<<1,      32,  0, stream>>>(A, bfv, last);
  viterbi_chunk_bp_g<<<NCHUNK, 32,  0, stream>>>(sent, trans, bfv, G);
  viterbi_scan_bwd  <<<1,      256, 0, stream>>>(G, last, exT);
  viterbi_emit      <<<NCHUNK, 32,  0, stream>>>(sent, trans, bfv, exT, out);
}